// StochasticAdvectionKernel_35579509080346
// MI455X (gfx1250) — compile-verified
//
#include <hip/hip_runtime.h>
#include <hip/hip_bf16.h>
#include <math.h>

// ---------------------------------------------------------------------------
// StochasticAdvectionKernel for MI455X (gfx1250, wave32).
//
// Output layout (f32, concatenated): tr[16,1024,1024] | drift[16,4,4,2] |
// disp[16,4,4,2,2] | scales[4,4] | mix[1].
//
// Roofline: 67 MB output write dominates (~3us @ 23.3 TB/s). We therefore
// write tr exactly once and RECOMPUTE the Gaussian blocks (cheap, rank-4
// WMMA) instead of materializing K twice. The quadratic form
//   quad = (u - v - d)^T inv (u - v - d)
// factors exactly into a K=4 matmul, so each 16x16 tile of quad is ONE
// v_wmma_f32_16x16x4_f32:
//   A[m,:] = [ (inv u_m)_0, (inv u_m)_1, u_m^T inv u_m, 1 ]
//   B[:,n] = [ -2 g_n0, -2 g_n1, 1, g_n^T inv g_n ],  g_n = v_n + d
// The three row-normalizations collapse to per-row scalars of the single
// row-sum S_K:  tr = ((1-mix)*K/(s1*s2) + mix*delta) / s3.
// (lags/centered sanitization clips at +-1e4 are no-ops for these magnitudes
// and are folded into the quad clip, which we keep faithful.)
// ---------------------------------------------------------------------------

#define T_DIM 16
#define C_DIM 4
#define S_DIM 256
#define CS    1024
#define TR_ELEMS   (16u * 1024u * 1024u)
#define DRIFT_OFF  16777216u
#define DISP_OFF   16777728u
#define SCALES_OFF 16778752u
#define MIX_OFF    16778768u
#define JIT        1e-5f
#define CLIPV      1e4f

typedef __attribute__((ext_vector_type(2))) float v2f;
typedef __attribute__((ext_vector_type(8))) float v8f;

__device__ __forceinline__ float sanf(float x) {
    // nan_to_num(nan=0, +-inf -> +-CLIP) followed by clip(+-CLIP)
    if (x != x) return 0.0f;
    return fminf(fmaxf(x, -CLIPV), CLIPV);
}

__device__ __forceinline__ float fix_log(float ld) {
    // jnp.nan_to_num(log, nan=0, posinf=20, neginf=-20): finite values kept.
    if (ld != ld) return 0.0f;
    if (__builtin_isinf(ld)) return ld > 0.0f ? 20.0f : -20.0f;
    return ld;
}

// ---------------------------------------------------------------------------
// Pass 1: per-(t,i,j) parameters. 256 blocks -> 256 threads, one block.
// Writes drift/disp/scales/mix tails of d_out, and 8-float param records + mix
// into ws (only ~8.2 KB of scratch used).
// ---------------------------------------------------------------------------
__global__ void prep_kernel(const float* __restrict__ means,
                            const float* __restrict__ covs,
                            const float* __restrict__ bsr,
                            const float* __restrict__ logit,
                            float* __restrict__ out,
                            float* __restrict__ ws) {
    int tid = threadIdx.x;
    if (tid >= 256) return;
    int t = tid >> 4, i = (tid >> 2) & 3, j = tid & 3;

    // drift = san(DT * (san(means[t,i]) - san(means[t,j])))
    const float* mi = means + (t * 4 + i) * 2;
    const float* mj = means + (t * 4 + j) * 2;
    float d0 = sanf(0.05f * (sanf(mi[0]) - sanf(mj[0])));
    float d1 = sanf(0.05f * (sanf(mi[1]) - sanf(mj[1])));
    out[DRIFT_OFF + tid * 2 + 0] = d0;
    out[DRIFT_OFF + tid * 2 + 1] = d1;

    // covs sanitized + symmetrized
    const float* ci = covs + (t * 4 + i) * 4;
    const float* cj = covs + (t * 4 + j) * 4;
    float ci00 = sanf(ci[0]), ci01 = 0.5f * (sanf(ci[1]) + sanf(ci[2])), ci11 = sanf(ci[3]);
    float cj00 = sanf(cj[0]), cj01 = 0.5f * (sanf(cj[1]) + sanf(cj[2])), cj11 = sanf(cj[3]);

    const float k2 = 2.0f * 0.05f * 0.05f;  // 2*DT^2
    float m00 = sanf(1.0f + k2 * (ci00 + cj00) + JIT);
    float m01 = sanf(k2 * (ci01 + cj01));
    float m11 = sanf(1.0f + k2 * (ci11 + cj11) + JIT);

    // _stable_inv_logdet: jitter escalation, first success wins.
    float ds = fmaxf(0.5f * (fabsf(m00) + fabsf(m11)), 1.0f);
    float j0 = fmaxf(JIT, 1e-6f);  // 1e-5
    float jit = j0;
    float ra = 0.f, rb = 0.f, rc = 0.f, det = 0.f;
    bool ok = false;
    for (int k = 0; k < 8; ++k) {
        float a = sanf(m00 + jit * ds);
        float b = sanf(m01);
        float c = sanf(m11 + jit * ds);
        float dt = a * c - b * b;
        if (!ok && a > 0.0f && dt > 0.0f) { ok = true; ra = a; rb = b; rc = c; det = dt; }
        jit *= 10.0f;
    }
    float iv00, iv01, iv11, ld;
    if (ok) {
        iv00 = sanf(rc / det);
        iv01 = sanf(-rb / det);
        iv11 = sanf(ra / det);
        ld = fix_log(logf(det));
    } else {
        // diagonal repair fallback
        float jf = j0 * 1e8f;
        float sd0 = (m00 != m00) ? 1.0f : m00;
        float sd1 = (m11 != m11) ? 1.0f : m11;
        sd0 = fmaxf(fabsf(sd0), jf * ds + 1e-4f);
        sd1 = fmaxf(fabsf(sd1), jf * ds + 1e-4f);
        ra = sanf(sd0 + jf * ds); rb = 0.0f; rc = sanf(sd1 + jf * ds);
        float dg0 = fmaxf(ra, 1e-6f), dg1 = fmaxf(rc, 1e-6f);
        iv00 = sanf(1.0f / dg0); iv01 = 0.0f; iv11 = sanf(1.0f / dg1);
        ld = fix_log(logf(dg0) + logf(dg1));
    }
    out[DISP_OFF + tid * 4 + 0] = ra;
    out[DISP_OFF + tid * 4 + 1] = rb;
    out[DISP_OFF + tid * 4 + 2] = rb;
    out[DISP_OFF + tid * 4 + 3] = rc;

    // scale = softplus(bsr[i,j]) + MIN_BLOCK_SCALE
    float x = bsr[i * 4 + j];
    float sc = fmaxf(x, 0.0f) + log1pf(expf(-fabsf(x))) + 0.01f;

    float* p = ws + tid * 8;
    p[0] = iv00; p[1] = iv01; p[2] = iv11;
    p[3] = d0;   p[4] = d1;   p[5] = ld; p[6] = sc; p[7] = 0.0f;

    if (tid < 16) out[SCALES_OFF + tid] = sc;  // t=0 threads: bsr index == tid
    if (tid == 0) {
        float mx = 1.0f / (1.0f + expf(-logit[0]));
        out[MIX_OFF] = mx;
        ws[2048] = mx;
    }
}

// Compute one 16x16 quad tile for column-tile bt via a single rank-4 WMMA.
__device__ __forceinline__ v8f quad_tile(v2f A, int bt, int row, int half,
                                         float i00, float i01, float i11,
                                         float d0, float d1,
                                         const float* __restrict__ sites) {
    int b_idx = bt * 16 + row;
    float vx = sites[2 * b_idx], vy = sites[2 * b_idx + 1];
    float gx = vx + d0, gy = vy + d1;          // g = v + drift  (c = u - g)
    float b0v = i00 * gx + i01 * gy;
    float b1v = i01 * gx + i11 * gy;
    float qb = gx * b0v + gy * b1v;            // g^T inv g
    v2f B;                                     // rows K0/K2 in .x, K1/K3 in .y
    B.x = half ? 1.0f : -2.0f * gx;
    B.y = half ? qb   : -2.0f * gy;
    v8f cz = {};
    return __builtin_amdgcn_wmma_f32_16x16x4_f32(false, A, false, B, (short)0,
                                                 cz, false, false);
}

// ---------------------------------------------------------------------------
// Pass 2: one workgroup per (t, i, 16-row strip); wave j covers its 256
// columns with 16 WMMA tiles. Row sums reduced via lane shuffles + LDS
// (no float atomics -> deterministic), then tiles are recomputed and the
// fully-normalized tr is written in a single pass over the 67 MB output.
// ---------------------------------------------------------------------------
__global__ void __launch_bounds__(128)
main_kernel(const float* __restrict__ sites,
            const float* __restrict__ ws,
            float* __restrict__ out) {
    __shared__ float rs[16][4];

    int wg = blockIdx.x;
    int t = wg >> 6;
    int i = (wg >> 4) & 3;
    int atile = wg & 15;
    int j = threadIdx.x >> 5;       // wave id = component column block
    int lane = threadIdx.x & 31;
    int row = lane & 15;
    int half = lane >> 4;

    const float* p = ws + (((t * 4 + i) * 4 + j) * 8);
    float i00 = p[0], i01 = p[1], i11 = p[2];
    float d0 = p[3], d1 = p[4];
    float nld = -0.5f * p[5];
    float sc = p[6];
    float mixv = ws[2048];

    // A operand: lanes 0-15 carry K0/K1 = inv*u, lanes 16-31 carry K2/K3 = [qa, 1]
    int a_idx = atile * 16 + row;
    float ux = sites[2 * a_idx], uy = sites[2 * a_idx + 1];
    float a0v = i00 * ux + i01 * uy;
    float a1v = i01 * ux + i11 * uy;
    float qa = ux * a0v + uy * a1v;
    v2f A;
    A.x = half ? qa : a0v;
    A.y = half ? 1.0f : a1v;

    float acc[8];
#pragma unroll
    for (int r = 0; r < 8; ++r) acc[r] = 0.0f;

    // --- pass A: row sums over this wave's 256 columns -------------------
    for (int bt = 0; bt < 16; ++bt) {
        v8f D = quad_tile(A, bt, row, half, i00, i01, i11, d0, d1, sites);
#pragma unroll
        for (int r = 0; r < 8; ++r) {
            float q = fminf(fmaxf(D[r], 0.0f), 60.0f);       // clip/nan_to_num
            float e = fminf(fmaxf(nld - q, -60.0f), 20.0f);
            acc[r] += sc * __expf(e);
        }
    }
    // reduce across the 16 lanes of each half (xor bits 0..3 stay in-group)
#pragma unroll
    for (int r = 0; r < 8; ++r) {
        float s = acc[r];
        s += __shfl_xor(s, 1);
        s += __shfl_xor(s, 2);
        s += __shfl_xor(s, 4);
        s += __shfl_xor(s, 8);
        acc[r] = s;
    }
    if (row == 0) {
#pragma unroll
        for (int r = 0; r < 8; ++r) rs[r + 8 * half][j] = acc[r];
    }
    __syncthreads();

    // per-row normalization scalars from S_K (deterministic fixed-order sum)
    float fmul[8], fdiv[8];
#pragma unroll
    for (int r = 0; r < 8; ++r) {
        int M = r + 8 * half;
        float SK = rs[M][0] + rs[M][1] + rs[M][2] + rs[M][3];
        float s1 = fmaxf(SK, JIT);
        float S2 = SK / s1;
        float s2 = fmaxf(S2, JIT);
        float S3 = (1.0f - mixv) * (S2 / s2) + mixv;
        float s3 = fmaxf(S3, JIT);
        fmul[r] = (1.0f - mixv) / (s1 * s2);
        fdiv[r] = 1.0f / s3;
    }

    // --- pass B: recompute tiles (compute is free) and write normalized tr
    size_t outBase = (size_t)t * (size_t)(CS * CS);
    for (int bt = 0; bt < 16; ++bt) {
        v8f D = quad_tile(A, bt, row, half, i00, i01, i11, d0, d1, sites);
#pragma unroll
        for (int r = 0; r < 8; ++r) {
            float q = fminf(fmaxf(D[r], 0.0f), 60.0f);
            float e = fminf(fmaxf(nld - q, -60.0f), 20.0f);
            float val = sc * __expf(e);
            int M = r + 8 * half;
            float o = val * fmul[r];
            if (j == i && bt == atile && row == M) o += mixv;  // identity mix
            o *= fdiv[r];
            int grow = i * S_DIM + atile * 16 + M;
            int gcol = j * S_DIM + bt * 16 + row;
            out[outBase + (size_t)grow * CS + (size_t)gcol] = o;
        }
    }
}

extern "C" void kernel_launch(void* const* d_in, const int* in_sizes, int n_in,
                              void* d_out, int out_size, void* d_ws, size_t ws_size,
                              hipStream_t stream) {
    const float* means = (const float*)d_in[0];
    const float* covs  = (const float*)d_in[1];
    const float* sites = (const float*)d_in[2];
    const float* bsr   = (const float*)d_in[3];
    const float* logit = (const float*)d_in[4];
    float* out = (float*)d_out;
    float* ws  = (float*)d_ws;

    prep_kernel<<<1, 256, 0, stream>>>(means, covs, bsr, logit, out, ws);
    main_kernel<<<T_DIM * C_DIM * (S_DIM / 16), 128, 0, stream>>>(sites, ws, out);
}